// GAT_15556371546776
// MI455X (gfx1250) — compile-verified
//
#include <hip/hip_runtime.h>

typedef float v2f __attribute__((ext_vector_type(2)));
typedef float v8f __attribute__((ext_vector_type(8)));

#define CO 64
#define LEAKY 0.2f

// ---- monotonic float <-> ordered-uint mapping (for atomic segment max) ----
__device__ __forceinline__ unsigned ford(float f) {
    unsigned u = __float_as_uint(f);
    return (u & 0x80000000u) ? ~u : (u | 0x80000000u);
}
__device__ __forceinline__ float funord(unsigned u) {
    return (u & 0x80000000u) ? __uint_as_float(u & 0x7FFFFFFFu)
                             : __uint_as_float(~u);
}

// ============================================================================
// WMMA GEMM: H[n_rows x 64] = act(X)[n_rows x CI] @ W[CI x 64]
// 256 threads = 8 waves; each wave produces a 16-row x 64-col strip using
// four 16x16 f32 accumulators and V_WMMA_F32_16X16X4_F32 (K step = 4).
// No divergent control flow before the WMMAs (EXEC must stay all-ones).
// ============================================================================
__global__ void __launch_bounds__(256)
gat_gemm_wmma(const float* __restrict__ X, const float* __restrict__ W,
              float* __restrict__ H, int n_rows, int CI, int relu_in)
{
    const int lane  = threadIdx.x & 31;
    const int wave  = threadIdx.x >> 5;
    const int m     = lane & 15;     // A row within tile / B+C column
    const int khalf = lane >> 4;     // 0 or 1 (selects K pair)
    const int tile_row = blockIdx.x * 128 + wave * 16;
    const int row      = tile_row + m;
    const bool rv      = row < n_rows;
    const float* xrow  = X + (size_t)(rv ? row : 0) * CI;

    v8f acc0 = {}; v8f acc1 = {}; v8f acc2 = {}; v8f acc3 = {};

    for (int k0 = 0; k0 < CI; k0 += 4) {
        const int k = k0 + 2 * khalf;
        float ax = xrow[k];
        float ay = xrow[k + 1];
        if (relu_in) { ax = fmaxf(ax, 0.f); ay = fmaxf(ay, 0.f); }
        if (!rv)     { ax = 0.f;            ay = 0.f;            }
        v2f a; a.x = ax; a.y = ay;

        const float* w0 = W + (size_t)k * CO + m;        // row k   of W
        const float* w1 = W + (size_t)(k + 1) * CO + m;  // row k+1 of W
        v2f b;
        b.x = w0[0];  b.y = w1[0];
        acc0 = __builtin_amdgcn_wmma_f32_16x16x4_f32(false, a, false, b, (short)0, acc0, false, false);
        b.x = w0[16]; b.y = w1[16];
        acc1 = __builtin_amdgcn_wmma_f32_16x16x4_f32(false, a, false, b, (short)0, acc1, false, false);
        b.x = w0[32]; b.y = w1[32];
        acc2 = __builtin_amdgcn_wmma_f32_16x16x4_f32(false, a, false, b, (short)0, acc2, false, false);
        b.x = w0[48]; b.y = w1[48];
        acc3 = __builtin_amdgcn_wmma_f32_16x16x4_f32(false, a, false, b, (short)0, acc3, false, false);
    }

    // C/D layout: VGPR r -> M = r + 8*khalf, N = lane&15
    const int srow_base = tile_row + 8 * khalf;
#pragma unroll
    for (int r = 0; r < 8; ++r) {
        const int srow = srow_base + r;
        if (srow < n_rows) {
            float* o = H + (size_t)srow * CO + m;
            o[0]  = acc0[r];
            o[16] = acc1[r];
            o[32] = acc2[r];
            o[48] = acc3[r];
        }
    }
}

// ============================================================================
// Per-node attention scores: s[n] = h[n].a_src, d[n] = h[n].a_dst
// One wave (32 lanes) per node; each lane covers 2 of 64 channels.
// Also initializes the per-node max (ordered-uint 0 == -inf-ish) and denom.
// ============================================================================
__global__ void __launch_bounds__(256)
node_scores(const float* __restrict__ H,
            const float* __restrict__ a_src, const float* __restrict__ a_dst,
            float* __restrict__ s, float* __restrict__ d,
            unsigned* __restrict__ m_ord, float* __restrict__ denom, int n)
{
    const int lane = threadIdx.x & 31;
    const int wave = threadIdx.x >> 5;
    const int node = blockIdx.x * 8 + wave;
    if (node >= n) return;

    const float h0 = H[(size_t)node * CO + lane];
    const float h1 = H[(size_t)node * CO + lane + 32];
    float sv = h0 * a_src[lane] + h1 * a_src[lane + 32];
    float dv = h0 * a_dst[lane] + h1 * a_dst[lane + 32];
#pragma unroll
    for (int off = 16; off > 0; off >>= 1) {
        sv += __shfl_xor(sv, off, 32);
        dv += __shfl_xor(dv, off, 32);
    }
    if (lane == 0) {
        s[node]     = sv;
        d[node]     = dv;
        m_ord[node] = 0u;     // below ford(-inf): safe init
        denom[node] = 0.f;
    }
}

// out[n][c] = bias[c]  (scatter-adds land on top -> final "+ b" is free)
__global__ void init_out(float* __restrict__ out, const float* __restrict__ b, int total)
{
    const int i = blockIdx.x * blockDim.x + threadIdx.x;
    if (i < total) out[i] = b[i & (CO - 1)];
}

// Pass A: e = leaky_relu(s[src] + d[dst]); segment max via ordered-uint atomicMax
__global__ void edge_score_max(const int* __restrict__ src, const int* __restrict__ dst,
                               const float* __restrict__ s, const float* __restrict__ d,
                               float* __restrict__ score, unsigned* __restrict__ m_ord,
                               int n_edges, int n_nodes)
{
    const int e = blockIdx.x * blockDim.x + threadIdx.x;
    if (e >= n_edges + n_nodes) return;
    int u, v;
    if (e < n_edges) { u = src[e]; v = dst[e]; }
    else             { u = v = e - n_edges; }     // implicit self-loop
    float sc = s[u] + d[v];
    sc = sc > 0.f ? sc : LEAKY * sc;
    score[e] = sc;
    atomicMax(&m_ord[v], ford(sc));
}

// Pass B: w = exp(e - max[dst]); segment sum of w into denom[dst]
__global__ void edge_expsum(const int* __restrict__ dst,
                            float* __restrict__ score,
                            const unsigned* __restrict__ m_ord,
                            float* __restrict__ denom,
                            int n_edges, int n_nodes)
{
    const int e = blockIdx.x * blockDim.x + threadIdx.x;
    if (e >= n_edges + n_nodes) return;
    const int v = (e < n_edges) ? dst[e] : (e - n_edges);
    const float w = expf(score[e] - funord(m_ord[v]));
    score[e] = w;
    atomicAdd(&denom[v], w);
}

// Pass C: out[dst] += (w/denom[dst]) * h[src]   (wave per edge, 2 ch/lane)
__global__ void __launch_bounds__(256)
edge_accum(const int* __restrict__ src, const int* __restrict__ dst,
           const float* __restrict__ score, const float* __restrict__ denom,
           const float* __restrict__ H, float* __restrict__ out,
           int n_edges, int n_nodes)
{
    const int lane = threadIdx.x & 31;
    const int wave = threadIdx.x >> 5;
    const int e = blockIdx.x * 8 + wave;
    if (e >= n_edges + n_nodes) return;
    int u, v;
    if (e < n_edges) { u = src[e]; v = dst[e]; }
    else             { u = v = e - n_edges; }
    const float alpha = score[e] / denom[v];
    const float h0 = H[(size_t)u * CO + lane];
    const float h1 = H[(size_t)u * CO + lane + 32];
    atomicAdd(&out[(size_t)v * CO + lane],      alpha * h0);
    atomicAdd(&out[(size_t)v * CO + lane + 32], alpha * h1);
}

__global__ void sigmoid_k(float* __restrict__ o, int total)
{
    const int i = blockIdx.x * blockDim.x + threadIdx.x;
    if (i < total) o[i] = 1.f / (1.f + expf(-o[i]));
}

// ============================================================================
extern "C" void kernel_launch(void* const* d_in, const int* in_sizes, int n_in,
                              void* d_out, int out_size, void* d_ws, size_t ws_size,
                              hipStream_t stream)
{
    const float* x  = (const float*)d_in[0];
    const int*   ei = (const int*)d_in[1];
    const float* W[3]    = {(const float*)d_in[2], (const float*)d_in[6],  (const float*)d_in[10]};
    const float* asrc[3] = {(const float*)d_in[3], (const float*)d_in[7],  (const float*)d_in[11]};
    const float* adst[3] = {(const float*)d_in[4], (const float*)d_in[8],  (const float*)d_in[12]};
    const float* bias[3] = {(const float*)d_in[5], (const float*)d_in[9],  (const float*)d_in[13]};

    const int n   = in_sizes[0] / 128;   // 100000 nodes
    const int ne  = in_sizes[1] / 2;     // 1600000 edges
    const int tot = ne + n;              // + self-loops
    const int* srcI = ei;
    const int* dstI = ei + ne;

    // workspace carve-out (~87 MB)
    char* p = (char*)d_ws;
    float*    Hb    = (float*)p;    p += (size_t)n * CO * sizeof(float);
    float*    featA = (float*)p;    p += (size_t)n * CO * sizeof(float);
    float*    featB = (float*)p;    p += (size_t)n * CO * sizeof(float);
    float*    sbuf  = (float*)p;    p += (size_t)n * sizeof(float);
    float*    dbuf  = (float*)p;    p += (size_t)n * sizeof(float);
    unsigned* mo    = (unsigned*)p; p += (size_t)n * sizeof(unsigned);
    float*    den   = (float*)p;    p += (size_t)n * sizeof(float);
    float*    sc    = (float*)p;    p += (size_t)tot * sizeof(float);
    (void)ws_size; (void)n_in; (void)out_size;

    const dim3 blk(256);
    const int nco = n * CO;

    auto run_layer = [&](const float* feat, int CI, int relu_in, int li, float* outbuf) {
        gat_gemm_wmma <<<(n + 127) / 128,   blk, 0, stream>>>(feat, W[li], Hb, n, CI, relu_in);
        node_scores   <<<(n + 7) / 8,       blk, 0, stream>>>(Hb, asrc[li], adst[li], sbuf, dbuf, mo, den, n);
        init_out      <<<(nco + 255) / 256, blk, 0, stream>>>(outbuf, bias[li], nco);
        edge_score_max<<<(tot + 255) / 256, blk, 0, stream>>>(srcI, dstI, sbuf, dbuf, sc, mo, ne, n);
        edge_expsum   <<<(tot + 255) / 256, blk, 0, stream>>>(dstI, sc, mo, den, ne, n);
        edge_accum    <<<(tot + 7) / 8,     blk, 0, stream>>>(srcI, dstI, sc, den, Hb, outbuf, ne, n);
    };

    float* outF = (float*)d_out;
    run_layer(x,     128, 0, 0, featA);   // layer 0 (fp32 in, 128 -> 64)
    run_layer(featA,  64, 1, 1, featB);   // layer 1 (relu fused into A-load)
    run_layer(featB,  64, 1, 2, outF);    // layer 2
    sigmoid_k<<<(nco + 255) / 256, blk, 0, stream>>>(outF, nco);
}